// ProposalLayer_15238543966349
// MI455X (gfx1250) — compile-verified
//
#include <hip/hip_runtime.h>
#include <stdint.h>

#define NBATCH   4
#define PRE_NMS  6000
#define PROPS    1000
#define NMS_THR  0.7f
#define NBINS    4096
#define CAP      8192
#define SORT_N   8192
#define KEEPW    188   // ceil(6000/32)

// ---- CDNA5 async global->LDS path (guarded; falls back to plain copy) ----
#if defined(__gfx1250__) && __has_builtin(__builtin_amdgcn_global_load_async_to_lds_b128)
#define HAVE_ASYNC 1
#else
#define HAVE_ASYNC 0
#endif

#if HAVE_ASYNC
typedef __attribute__((__vector_size__(4 * sizeof(int)))) int v4i;
typedef __attribute__((address_space(1))) v4i* gptr_v4i;   // global (AS1)
typedef __attribute__((address_space(3))) v4i* lptr_v4i;   // LDS (AS3)
__device__ __forceinline__ gptr_v4i to_g(const void* p) {
  return (gptr_v4i)(uintptr_t)p;                 // global VA identical in AS1
}
__device__ __forceinline__ lptr_v4i to_l(const void* p) {
  return (lptr_v4i)(uint32_t)(uintptr_t)p;       // low 32 bits of generic LDS ptr = LDS offset
}
#endif

__device__ __forceinline__ float clip01(float v) { return fminf(fmaxf(v, 0.0f), 1.0f); }

// ---------------- K0: zero histogram + candidate counters ----------------
__global__ void k_zero(unsigned* __restrict__ hist, unsigned* __restrict__ cnt) {
  int t = blockIdx.x * blockDim.x + threadIdx.x;
  if (t < NBATCH * NBINS) hist[t] = 0u;
  if (t < NBATCH)         cnt[t]  = 0u;
}

// ---------------- K1: per-batch score histogram (LDS privatized) ----------
__global__ void k_hist(const float* __restrict__ scores, unsigned* __restrict__ ghist, int N) {
  __shared__ unsigned lh[NBINS];
  const int b = blockIdx.y;
  for (int i = threadIdx.x; i < NBINS; i += blockDim.x) lh[i] = 0u;
  __syncthreads();
  const float* s = scores + (size_t)b * (size_t)N * 2;
  const int base = blockIdx.x * blockDim.x * 8;
  for (int it = 0; it < 8; ++it) {
    int i = base + it * blockDim.x + threadIdx.x;
    if (i < N) {
      __builtin_prefetch(s + 2 * (i + 2048) + 1, 0, 0);   // global_prefetch_b8
      float v = s[2 * i + 1];
      int bin = (int)(v * (float)NBINS);
      bin = bin < 0 ? 0 : (bin > NBINS - 1 ? NBINS - 1 : bin);
      atomicAdd(&lh[bin], 1u);
    }
  }
  __syncthreads();
  unsigned* gh = ghist + b * NBINS;
  for (int i = threadIdx.x; i < NBINS; i += blockDim.x)
    if (lh[i]) atomicAdd(&gh[i], lh[i]);
}

// ---------------- K2: find cutoff bin so suffix count >= PRE_NMS ----------
__global__ void k_thresh(const unsigned* __restrict__ ghist, int* __restrict__ cutoff) {
  const int b = blockIdx.x;
  if (threadIdx.x == 0) {
    const unsigned* h = ghist + b * NBINS;
    unsigned acc = 0; int c = 0;
    for (int bin = NBINS - 1; bin >= 0; --bin) {
      acc += h[bin];
      if (acc >= PRE_NMS) { c = bin; break; }
    }
    cutoff[b] = c;
  }
}

// ---------------- K3: compact candidates (score,index) keys ---------------
__global__ void k_compact(const float* __restrict__ scores, const int* __restrict__ cutoff,
                          unsigned* __restrict__ cnt, unsigned long long* __restrict__ keys,
                          int N) {
  const int b = blockIdx.y;
  const int cut = cutoff[b];
  const float* s = scores + (size_t)b * (size_t)N * 2;
  unsigned long long* kb = keys + (size_t)b * CAP;
  const int base = blockIdx.x * blockDim.x * 8;
  for (int it = 0; it < 8; ++it) {
    int i = base + it * blockDim.x + threadIdx.x;
    if (i < N) {
      float v = s[2 * i + 1];
      int bin = (int)(v * (float)NBINS);
      bin = bin < 0 ? 0 : (bin > NBINS - 1 ? NBINS - 1 : bin);
      if (bin >= cut) {
        unsigned pos = atomicAdd(&cnt[b], 1u);
        if (pos < CAP) {
          unsigned sb = __float_as_uint(v);          // v >= 0 => bit order == float order
          kb[pos] = ((unsigned long long)sb << 32) | (unsigned)(~(unsigned)i);
        }
      }
    }
  }
}

// ------- K4: in-LDS bitonic sort (desc) + gather + box delta + clip -------
__global__ void __launch_bounds__(1024)
k_sort_box(const unsigned long long* __restrict__ keys, const unsigned* __restrict__ cnt,
           const float* __restrict__ deltas, const float* __restrict__ anchors,
           float* __restrict__ boxes, int N) {
  extern __shared__ unsigned long long sk[];   // SORT_N * 8 = 64 KB dynamic LDS
  const int b = blockIdx.x;
  const int tid = threadIdx.x;
  unsigned count = cnt[b]; if (count > CAP) count = CAP;
  const unsigned long long* kb = keys + (size_t)b * CAP;
  for (int i = tid; i < SORT_N; i += 1024) sk[i] = (i < (int)count) ? kb[i] : 0ULL;
  __syncthreads();
  for (unsigned k = 2; k <= SORT_N; k <<= 1) {
    for (unsigned j = k >> 1; j > 0; j >>= 1) {
      for (unsigned i = tid; i < SORT_N; i += 1024) {
        unsigned ixj = i ^ j;
        if (ixj > i) {
          unsigned long long a = sk[i], c = sk[ixj];
          bool doswap = ((i & k) == 0) ? (a < c) : (a > c);   // descending overall
          if (doswap) { sk[i] = c; sk[ixj] = a; }
        }
      }
      __syncthreads();
    }
  }
  const float4* anc = (const float4*)anchors + (size_t)b * N;
  const float4* del = (const float4*)deltas  + (size_t)b * N;
  float4* ob = (float4*)boxes + (size_t)b * PRE_NMS;
  for (int t = tid; t < PRE_NMS; t += 1024) {
    unsigned long long key = sk[t];
    float4 r = {0.f, 0.f, 0.f, 0.f};
    if (key != 0ULL) {
      unsigned idx = ~(unsigned)(key & 0xffffffffULL);
      float4 a = anc[idx];
      float4 d = del[idx];
      float w = a.z - a.x, h = a.w - a.y;
      float cx = a.x + 0.5f * w + d.x * 0.1f * w;
      float cy = a.y + 0.5f * h + d.y * 0.1f * h;
      w = w * expf(d.z * 0.2f);
      h = h * expf(d.w * 0.2f);
      r.x = clip01(cx - 0.5f * w);
      r.y = clip01(cy - 0.5f * h);
      r.z = clip01(cx + 0.5f * w);
      r.w = clip01(cy + 0.5f * h);
    }
    ob[t] = r;
  }
}

// ---------------- K5: LDS-resident greedy NMS (one block per batch) -------
__global__ void __launch_bounds__(1024)
k_nms(const float* __restrict__ boxes, float* __restrict__ out) {
  extern __shared__ float4 sbox[];            // PRE_NMS * 16 = 96 KB dynamic LDS
  __shared__ unsigned skeep[KEEPW];
  __shared__ int klist[PROPS];
  const int b = blockIdx.x;
  const int tid = threadIdx.x;
  const float4* gb = (const float4*)boxes + (size_t)b * PRE_NMS;
  for (int t = tid; t < PRE_NMS; t += 1024) {
#if HAVE_ASYNC
    __builtin_amdgcn_global_load_async_to_lds_b128(to_g(gb + t), to_l(sbox + t), 0, 0);
#else
    sbox[t] = gb[t];
#endif
  }
  for (int w = tid; w < KEEPW; w += 1024)
    skeep[w] = (w == KEEPW - 1) ? 0x0000FFFFu : 0xFFFFFFFFu;   // 6000 = 187*32 + 16
#if HAVE_ASYNC
#if __has_builtin(__builtin_amdgcn_s_wait_asynccnt)
  __builtin_amdgcn_s_wait_asynccnt(0);
#else
  asm volatile("s_wait_asynccnt 0" ::: "memory");
#endif
#endif
  __syncthreads();

  int kept = 0;                                // identical on every thread
  for (int i = 0; i < PRE_NMS; ++i) {
    if (!((skeep[i >> 5] >> (i & 31)) & 1u)) continue;   // uniform read
    if (tid == 0) klist[kept] = i;
    ++kept;
    if (kept >= PROPS) break;                  // first 1000 kept decided; stop
    float4 bi = sbox[i];
    float ai = (bi.z - bi.x) * (bi.w - bi.y);
    for (int j = i + 1 + tid; j < PRE_NMS; j += 1024) {
      if (!((skeep[j >> 5] >> (j & 31)) & 1u)) continue; // skip-work filter only
      float4 bj = sbox[j];
      float iw = fminf(bi.z, bj.z) - fmaxf(bi.x, bj.x);
      float ih = fminf(bi.w, bj.w) - fmaxf(bi.y, bj.y);
      iw = iw > 0.f ? iw : 0.f;
      ih = ih > 0.f ? ih : 0.f;
      float inter = iw * ih;
      float aj = (bj.z - bj.x) * (bj.w - bj.y);
      float iou = inter / (ai + aj - inter + 1e-12f);
      if (iou > NMS_THR) atomicAnd(&skeep[j >> 5], ~(1u << (j & 31)));
    }
    __syncthreads();
  }
  __syncthreads();

  const int kt = kept < PROPS ? kept : PROPS;
  float4* ob = (float4*)out + (size_t)b * PROPS;
  for (int s = tid; s < PROPS; s += 1024) {
    float4 v = {0.f, 0.f, 0.f, 0.f};
    if (s < kt) v = sbox[klist[s]];
    ob[s] = v;
  }
}

extern "C" void kernel_launch(void* const* d_in, const int* in_sizes, int n_in,
                              void* d_out, int out_size, void* d_ws, size_t ws_size,
                              hipStream_t stream) {
  (void)n_in; (void)out_size; (void)ws_size;
  const float* scores  = (const float*)d_in[0];   // (4, N, 2)
  const float* deltas  = (const float*)d_in[1];   // (4, N, 4)
  const float* anchors = (const float*)d_in[2];   // (4, N, 4)
  const int N = in_sizes[2] / (NBATCH * 4);

  char* ws = (char*)d_ws;
  unsigned*           hist   = (unsigned*)(ws);                        // 64 KB
  int*                cutoff = (int*)(ws + 65536);                     // 16 B
  unsigned*           cnt    = (unsigned*)(ws + 65568);                // 16 B
  unsigned long long* keys   = (unsigned long long*)(ws + 131072);     // 256 KB
  float*              boxes  = (float*)(ws + 131072 + 262144);         // 375 KB

  k_zero<<<(NBATCH * NBINS + 255) / 256, 256, 0, stream>>>(hist, cnt);
  dim3 g1((N + 2047) / 2048, NBATCH);
  k_hist<<<g1, 256, 0, stream>>>(scores, hist, N);
  k_thresh<<<NBATCH, 32, 0, stream>>>(hist, cutoff);
  k_compact<<<g1, 256, 0, stream>>>(scores, cutoff, cnt, keys, N);
  k_sort_box<<<NBATCH, 1024, SORT_N * sizeof(unsigned long long), stream>>>(
      keys, cnt, deltas, anchors, boxes, N);
  k_nms<<<NBATCH, 1024, PRE_NMS * sizeof(float4), stream>>>(boxes, (float*)d_out);
}